// SS2D_4956392259728
// MI455X (gfx1250) — compile-verified
//
#include <hip/hip_runtime.h>
#include <math.h>

// SS2D for MI455X (gfx1250): fp32 WMMA GEMMs + VALU selective scan with
// async memory->LDS staging (ASYNCcnt path).
#define B_   8
#define C_   192
#define H_   48
#define W_   48
#define L_   2304
#define DE_  384   // d_expand == d_inner
#define K_   4
#define N_   16
#define R_   12
#define CDBL_ 44   // R + 2N

typedef float v2f __attribute__((ext_vector_type(2)));
typedef float v8f __attribute__((ext_vector_type(8)));

__device__ __forceinline__ float gelu_exact(float x) {
  return 0.5f * x * (1.0f + erff(x * 0.70710678118654752f));
}

__device__ __forceinline__ void wait_asynccnt0() {
#if defined(__has_builtin)
#if __has_builtin(__builtin_amdgcn_s_wait_asynccnt)
  __builtin_amdgcn_s_wait_asynccnt(0);
#else
  asm volatile("s_wait_asynccnt 0x0" ::: "memory");
#endif
#else
  asm volatile("s_wait_asynccnt 0x0" ::: "memory");
#endif
}

// Async 4-byte memory->LDS copy, one element per active lane (ASYNCcnt).
__device__ __forceinline__ void async_load_f32_to_lds(const float* gaddr, float* lds_generic) {
  // Low 32 bits of a generic pointer to __shared__ are the workgroup-relative
  // LDS byte offset (aperture bits live in [63:32]); VDST takes that offset.
  unsigned int lds_off = (unsigned int)(unsigned long long)lds_generic;
  asm volatile("global_load_async_to_lds_b32 %0, %1, off"
               :: "v"(lds_off), "v"(gaddr) : "memory");
}

// Two adjacent 16x16 fp32 output tiles per wave (cols n0 and n0+16), sharing
// one A fragment per K-step: V_WMMA_F32_16X16X4_F32, wave32.
// A: row-major M x Kdim (lda); MASKED pads rows >= Mmax with zeros.
// Bm: row-major Kdim x N (ldb).
struct acc2 { v8f c0, c1; };

template <bool MASKED>
__device__ __forceinline__ acc2 wmma_f32_tile2(const float* __restrict__ A, int lda,
                                               int m0, int Mmax,
                                               const float* __restrict__ Bm, long ldb,
                                               int n0, int Kdim) {
  const int lane = threadIdx.x & 31;
  const int mr = lane & 15;            // A row / B col within tile
  const int ko = (lane >> 4) << 1;     // lanes 16-31 hold K+2, K+3
  int ar = m0 + mr;
  float amask = 1.0f;
  if (MASKED) {
    amask = (ar < Mmax) ? 1.0f : 0.0f;
    ar = (ar < Mmax - 1) ? ar : (Mmax - 1);
  }
  acc2 r; r.c0 = {}; r.c1 = {};
  for (int kk = 0; kk < Kdim; kk += 4) {
    v2f a, b0, b1;
    a.x = A[(size_t)ar * lda + kk + ko];
    a.y = A[(size_t)ar * lda + kk + ko + 1];
    if (MASKED) { a.x *= amask; a.y *= amask; }
    const float* brow0 = Bm + (size_t)(kk + ko) * ldb + n0 + mr;
    const float* brow1 = brow0 + ldb;
    b0.x = brow0[0];  b0.y = brow1[0];
    b1.x = brow0[16]; b1.y = brow1[16];
    r.c0 = __builtin_amdgcn_wmma_f32_16x16x4_f32(false, a, false, b0,
                                                 (short)0, r.c0, false, false);
    r.c1 = __builtin_amdgcn_wmma_f32_16x16x4_f32(false, a, false, b1,
                                                 (short)0, r.c1, false, false);
  }
  return r;
}

// K1: xz = in_proj_w (768x192) @ x (192 x L per batch); split -> xx_pre, z1=gelu(z)
__global__ void ss2d_inproj_wmma(const float* __restrict__ x, const float* __restrict__ w,
                                 float* __restrict__ xx_pre, float* __restrict__ z1) {
  const int n0 = (blockIdx.x * blockDim.y + threadIdx.y) * 32;
  const int m0 = blockIdx.y * 16;
  const int b  = blockIdx.z;
  const float* Bm = x + (size_t)b * C_ * L_;
  acc2 acc = wmma_f32_tile2<false>(w, C_, m0, 2 * DE_, Bm, L_, n0, C_);
  union { v8f v; float f[8]; } u0, u1; u0.v = acc.c0; u1.v = acc.c1;
  const int lane = threadIdx.x & 31;
  const int n  = n0 + (lane & 15);
  const int mh = (lane >> 4) << 3;
#pragma unroll
  for (int r = 0; r < 8; ++r) {
    int o = m0 + r + mh;
    if (o < DE_) {
      float* p = xx_pre + ((size_t)b * DE_ + o) * L_ + n;
      p[0] = u0.f[r]; p[16] = u1.f[r];
    } else {
      float* p = z1 + ((size_t)b * DE_ + (o - DE_)) * L_ + n;
      p[0] = gelu_exact(u0.f[r]); p[16] = gelu_exact(u1.f[r]);
    }
  }
}

// K2: depthwise 3x3 SAME conv + bias + gelu; also emits transposed (W,H) copy
__global__ void ss2d_dwconv_gelu(const float* __restrict__ xin, const float* __restrict__ cw,
                                 const float* __restrict__ cb,
                                 float* __restrict__ xconv, float* __restrict__ xt) {
  int idx = blockIdx.x * blockDim.x + threadIdx.x;
  if (idx >= B_ * DE_ * L_) return;
  int l = idx % L_; int c = (idx / L_) % DE_; int b = idx / (L_ * DE_);
  int h = l / W_, w = l % W_;
  const float* src = xin + ((size_t)b * DE_ + c) * L_;
  const float* k9  = cw + c * 9;
  float s = cb[c];
#pragma unroll
  for (int ky = 0; ky < 3; ++ky) {
    int hy = h + ky - 1;
    if ((unsigned)hy >= (unsigned)H_) continue;
#pragma unroll
    for (int kx = 0; kx < 3; ++kx) {
      int wx = w + kx - 1;
      if ((unsigned)wx >= (unsigned)W_) continue;
      s += k9[ky * 3 + kx] * src[hy * W_ + wx];
    }
  }
  s = gelu_exact(s);
  xconv[((size_t)b * DE_ + c) * L_ + h * W_ + w] = s;
  xt  [((size_t)b * DE_ + c) * L_ + w * H_ + h] = s;
}

// K3: x_dbl[b,k] = x_proj_w[k] (44x384) @ xs_k (384 x L). Directions 2/3 share
// the unflipped layouts of 0/1; the flip is folded into the scan's iteration order.
__global__ void ss2d_xdbl_wmma(const float* __restrict__ xpw, const float* __restrict__ xconv,
                               const float* __restrict__ xt, float* __restrict__ xdbl) {
  const int n0 = (blockIdx.x * blockDim.y + threadIdx.y) * 32;
  const int m0 = blockIdx.y * 16;
  const int b = blockIdx.z >> 2, k = blockIdx.z & 3;
  const float* A  = xpw + (size_t)k * CDBL_ * DE_;
  const float* Bm = ((k & 1) ? xt : xconv) + (size_t)b * DE_ * L_;
  acc2 acc = wmma_f32_tile2<true>(A, DE_, m0, CDBL_, Bm, L_, n0, DE_);
  union { v8f v; float f[8]; } u0, u1; u0.v = acc.c0; u1.v = acc.c1;
  const int lane = threadIdx.x & 31;
  const int n  = n0 + (lane & 15);
  const int mh = (lane >> 4) << 3;
  float* out = xdbl + (size_t)(b * K_ + k) * CDBL_ * L_;
#pragma unroll
  for (int r = 0; r < 8; ++r) {
    int m = m0 + r + mh;
    if (m < CDBL_) {
      float* p = out + (size_t)m * L_ + n;
      p[0] = u0.f[r]; p[16] = u1.f[r];
    }
  }
}

// K4: delta = softplus(dt_w[k] (384x12) @ dts (12 x L) + dt_b)
__global__ void ss2d_dt_wmma(const float* __restrict__ dtw, const float* __restrict__ dtb,
                             const float* __restrict__ xdbl, float* __restrict__ delta) {
  const int n0 = (blockIdx.x * blockDim.y + threadIdx.y) * 32;
  const int m0 = blockIdx.y * 16;
  const int b = blockIdx.z >> 2, k = blockIdx.z & 3;
  const float* A  = dtw + (size_t)k * DE_ * R_;
  const float* Bm = xdbl + (size_t)(b * K_ + k) * CDBL_ * L_;  // rows 0..11 = dts
  acc2 acc = wmma_f32_tile2<false>(A, R_, m0, DE_, Bm, L_, n0, R_);
  union { v8f v; float f[8]; } u0, u1; u0.v = acc.c0; u1.v = acc.c1;
  const int lane = threadIdx.x & 31;
  const int n  = n0 + (lane & 15);
  const int mh = (lane >> 4) << 3;
  float* out = delta + (size_t)(b * K_ + k) * DE_ * L_;
#pragma unroll
  for (int r = 0; r < 8; ++r) {
    int d = m0 + r + mh;
    float bia = dtb[k * DE_ + d];
    float v0 = u0.f[r] + bia;
    float v1 = u1.f[r] + bia;
    float* p = out + (size_t)d * L_ + n;
    p[0]  = (v0 > 20.0f) ? v0 : log1pf(expf(v0));
    p[16] = (v1 > 20.0f) ? v1 : log1pf(expf(v1));
  }
}

// K5: selective scan. One thread per (b,k,d) row, 16 states in VGPRs.
// B_t/C_t staged with GLOBAL_LOAD_ASYNC_TO_LDS_B32 (ASYNCcnt) one step ahead,
// double-buffered, one barrier per step. ys overwrites delta in place.
// k>=2 iterates backward over the stored (unflipped) sequence.
__global__ void ss2d_scan(const float* __restrict__ A_logs, const float* __restrict__ Ds,
                          const float* __restrict__ xconv, const float* __restrict__ xt,
                          const float* __restrict__ xdbl, float* __restrict__ dy) {
  const int b = blockIdx.x >> 2, k = blockIdx.x & 3;
  const int d = blockIdx.y * blockDim.x + threadIdx.x;
  const bool fwd = (k < 2);
  float Ar[N_];
#pragma unroll
  for (int n = 0; n < N_; ++n)
    Ar[n] = -__expf(A_logs[(size_t)(k * DE_ + d) * N_ + n]);
  const float dsk = Ds[k * DE_ + d];
  const float* u_src = ((k & 1) ? xt : xconv) + ((size_t)b * DE_ + d) * L_;
  const float* bc = xdbl + (size_t)(b * K_ + k) * CDBL_ * L_;
  float* dyrow = dy + ((size_t)(b * K_ + k) * DE_ + d) * L_;
  float h[N_];
#pragma unroll
  for (int n = 0; n < N_; ++n) h[n] = 0.0f;

  // sBC[pb][0..15] = B_t, sBC[pb][16..31] = C_t
  __shared__ float sBC[2][32];

  // Prologue: async-stage step 0 (lanes of wave 0, one float each).
  if (threadIdx.x < 32) {
    const int i = threadIdx.x;
    async_load_f32_to_lds(bc + (size_t)(R_ + i) * L_ + (fwd ? 0 : (L_ - 1)),
                          &sBC[0][i]);
  }

  for (int t = 0; t < L_; ++t) {
    const int s  = fwd ? t : (L_ - 1 - t);
    const int pb = t & 1;
    if (threadIdx.x < 32) wait_asynccnt0();   // data for step t landed in LDS
    __syncthreads();                          // publish to all 4 waves
    if (threadIdx.x < 32 && (t + 1) < L_) {   // prefetch step t+1 into other buf
      const int i  = threadIdx.x;
      const int s1 = fwd ? (t + 1) : (L_ - 2 - t);
      async_load_f32_to_lds(bc + (size_t)(R_ + i) * L_ + s1, &sBC[pb ^ 1][i]);
    }
    int tp = t + 8;
    if (tp < L_) {
      int sp = fwd ? tp : (L_ - 1 - tp);
      __builtin_prefetch((const void*)(dyrow + sp), 0, 1);   // global_prefetch_b8
      __builtin_prefetch((const void*)(u_src + sp), 0, 1);
    }
    const float dl = dyrow[s];
    const float uu = u_src[s];
    const float du = dl * uu;
    float y = 0.0f;
#pragma unroll
    for (int n = 0; n < N_; ++n) {
      const float dA = __expf(dl * Ar[n]);   // v_exp_f32 co-executes w/ FMA
      h[n] = h[n] * dA + du * sBC[pb][n];
      y += h[n] * sBC[pb][16 + n];
    }
    dyrow[s] = y + uu * dsk;                 // in-place ys
  }
}

// K6: CrossMerge (4-way sum, transpose map for k=1,3) + LayerNorm over D + gate
__global__ void ss2d_merge_ln_gate(const float* __restrict__ ybuf, const float* __restrict__ z1,
                                   const float* __restrict__ lnw, const float* __restrict__ lnb,
                                   float* __restrict__ yfin) {
  const int blk = blockIdx.x;
  const int b = blk / L_, l = blk % L_;
  const int h = l / W_, w = l % W_;
  const int lc = w * H_ + h;
  const int tid = threadIdx.x;
  float vals[3];
  float s1 = 0.0f, s2 = 0.0f;
#pragma unroll
  for (int j = 0; j < 3; ++j) {
    int d = tid + j * 128;
    float v = ybuf[((size_t)(b * K_ + 0) * DE_ + d) * L_ + l]
            + ybuf[((size_t)(b * K_ + 2) * DE_ + d) * L_ + l]
            + ybuf[((size_t)(b * K_ + 1) * DE_ + d) * L_ + lc]
            + ybuf[((size_t)(b * K_ + 3) * DE_ + d) * L_ + lc];
    vals[j] = v; s1 += v; s2 += v * v;
  }
  __shared__ float r1[128], r2[128];
  r1[tid] = s1; r2[tid] = s2;
  __syncthreads();
  for (int off = 64; off > 0; off >>= 1) {
    if (tid < off) { r1[tid] += r1[tid + off]; r2[tid] += r2[tid + off]; }
    __syncthreads();
  }
  const float mu   = r1[0] * (1.0f / DE_);
  const float var  = r2[0] * (1.0f / DE_) - mu * mu;
  const float rinv = rsqrtf(var + 1e-5f);
#pragma unroll
  for (int j = 0; j < 3; ++j) {
    int d = tid + j * 128;
    float v = (vals[j] - mu) * rinv * lnw[d] + lnb[d];
    v *= z1[((size_t)b * DE_ + d) * L_ + l];
    yfin[((size_t)b * DE_ + d) * L_ + l] = v;
  }
}

// K7: out = out_proj_w (192x384) @ yfin (384 x L per batch)
__global__ void ss2d_outproj_wmma(const float* __restrict__ w, const float* __restrict__ yfin,
                                  float* __restrict__ out) {
  const int n0 = (blockIdx.x * blockDim.y + threadIdx.y) * 32;
  const int m0 = blockIdx.y * 16;
  const int b  = blockIdx.z;
  const float* Bm = yfin + (size_t)b * DE_ * L_;
  acc2 acc = wmma_f32_tile2<false>(w, DE_, m0, C_, Bm, L_, n0, DE_);
  union { v8f v; float f[8]; } u0, u1; u0.v = acc.c0; u1.v = acc.c1;
  const int lane = threadIdx.x & 31;
  const int n  = n0 + (lane & 15);
  const int mh = (lane >> 4) << 3;
#pragma unroll
  for (int r = 0; r < 8; ++r) {
    int c = m0 + r + mh;
    float* p = out + ((size_t)b * C_ + c) * L_ + n;
    p[0] = u0.f[r]; p[16] = u1.f[r];
  }
}

extern "C" void kernel_launch(void* const* d_in, const int* in_sizes, int n_in,
                              void* d_out, int out_size, void* d_ws, size_t ws_size,
                              hipStream_t stream) {
  const float* x    = (const float*)d_in[0];
  const float* ipw  = (const float*)d_in[1];
  const float* cw   = (const float*)d_in[2];
  const float* cb   = (const float*)d_in[3];
  const float* xpw  = (const float*)d_in[4];
  const float* dtw  = (const float*)d_in[5];
  const float* dtb  = (const float*)d_in[6];
  const float* alog = (const float*)d_in[7];
  const float* dsv  = (const float*)d_in[8];
  const float* lnw  = (const float*)d_in[9];
  const float* lnb  = (const float*)d_in[10];
  const float* opw  = (const float*)d_in[11];
  float* out = (float*)d_out;
  float* ws  = (float*)d_ws;

  const size_t SZ = (size_t)B_ * DE_ * L_;             // 7,077,888 floats
  float* xx_pre = ws;                                  // dead after K2 -> reused as yfin
  float* z1     = ws + SZ;
  float* xconv  = ws + 2 * SZ;
  float* xt     = ws + 3 * SZ;
  float* xdbl   = ws + 4 * SZ;                         // B*K*44*L
  float* delta  = xdbl + (size_t)B_ * K_ * CDBL_ * L_; // B*K*384*L (ys in place)
  float* yfin   = xx_pre;

  dim3 blk32(32, 4);  // 4 wave32s per block, two 16x16 tiles each (32 cols/wave)
  ss2d_inproj_wmma<<<dim3(L_ / 128, (2 * DE_) / 16, B_), blk32, 0, stream>>>(x, ipw, xx_pre, z1);
  int tot = B_ * DE_ * L_;
  ss2d_dwconv_gelu<<<(tot + 255) / 256, 256, 0, stream>>>(xx_pre, cw, cb, xconv, xt);
  ss2d_xdbl_wmma<<<dim3(L_ / 128, 3, B_ * K_), blk32, 0, stream>>>(xpw, xconv, xt, xdbl);
  ss2d_dt_wmma<<<dim3(L_ / 128, DE_ / 16, B_ * K_), blk32, 0, stream>>>(dtw, dtb, xdbl, delta);
  ss2d_scan<<<dim3(B_ * K_, DE_ / 128), 128, 0, stream>>>(alog, dsv, xconv, xt, xdbl, delta);
  ss2d_merge_ln_gate<<<B_ * L_, 128, 0, stream>>>(delta, z1, lnw, lnb, yfin);
  ss2d_outproj_wmma<<<dim3(L_ / 128, C_ / 16, B_), blk32, 0, stream>>>(opw, yfin, out);
}